// KHDistortionDSP_54571854463259
// MI455X (gfx1250) — compile-verified
//
#include <hip/hip_runtime.h>
#include <math.h>

// CDNA5 / gfx1250: scan-as-matmul high-pass IIR using V_WMMA_F32_16X16X4_F32.
typedef float v2f __attribute__((ext_vector_type(2)));
typedef float v8f __attribute__((ext_vector_type(8)));

#define HPF_ALPHA 0.99715430522096418f
#define WAVES 8
#define TILE  2048   /* WAVES * 256 samples per macro-tile */

__device__ __forceinline__ float waveshape(float p, int t) {
  if (t == 0) return tanhf(p);
  if (t == 1) return fminf(1.0f, fmaxf(-1.0f, p));
  if (t == 2) { float r = p + 1.0f; r -= 4.0f * floorf(r * 0.25f); return fabsf(r - 2.0f) - 1.0f; }
  if (t == 3) return sinf(p * 1.5707963267948966f);
  return p / (1.0f + fabsf(p));
}

__global__ __launch_bounds__(256)
void KHDistortionDSP_hpf_wmma(const float* __restrict__ X,
                              const float* __restrict__ drive,
                              const float* __restrict__ bias,
                              const float* __restrict__ mix,
                              const int*   __restrict__ typep,
                              float* __restrict__ OUT,
                              int N, int segLen, int warm) {
  __shared__ float dbuf[WAVES][256];   // d-values for B-fragment gather (wave-private)
  __shared__ float ybuf[WAVES][256];   // layout transpose for coalesced stores
  __shared__ float waveB[WAVES];       // per-wave affine carry aggregates

  const int row  = blockIdx.y;
  const int seg  = blockIdx.x;
  const int tid  = threadIdx.x;
  const int wid  = tid >> 5;
  const int lane = tid & 31;
  const int half = lane >> 4;          // 0: lanes 0-15, 1: lanes 16-31
  const int j    = lane & 15;          // column / row-in-half index
  const int t    = *typep;

  const float* xrow = X   + (size_t)row * (size_t)N;
  float*       orow = OUT + (size_t)row * (size_t)N;
  const float g  = powf(10.0f, drive[row] * 2.4f);
  const float dc = bias[row] * 2.0f - 1.0f;
  const float mv = mix[row];
  const float om = 1.0f - mv;

  // A fragments: T[m][k] = alpha^(m-k) lower-triangular, sliced into four 16x4 K-chunks.
  // f32 A layout: lanes 0-15 hold (M=lane, K=0),(K=1); lanes 16-31 hold (M=lane-16, K=2),(K=3).
  v2f afrag[4];
#pragma unroll
  for (int kk = 0; kk < 4; ++kk) {
    int k0 = 4 * kk + (half ? 2 : 0);
    int k1 = k0 + 1;
    afrag[kk].x = (j >= k0) ? powf(HPF_ALPHA, (float)(j - k0)) : 0.0f;
    afrag[kk].y = (j >= k1) ? powf(HPF_ALPHA, (float)(j - k1)) : 0.0f;
  }
  // Carry-injection coefficients: C/D VGPR v holds row M = v + 8*half -> alpha^(M+1).
  float pw[8];
#pragma unroll
  for (int v = 0; v < 8; ++v) pw[v] = powf(HPF_ALPHA, (float)(v + 1 + 8 * half));
  const float m16  = powf(HPF_ALPHA, 16.0f);   // per-16-chunk carry multiplier
  const float a256 = powf(HPF_ALPHA, 256.0f);  // per-wave-block carry multiplier

  const int w0       = seg ? warm : 0;
  const int start0   = seg * segLen - w0;      // includes warm-up region
  const int outStart = seg * segLen;
  const int nTiles   = (segLen + w0) / TILE;

  float cMacro = 0.0f;                         // running carry across macro-tiles

  for (int tile = 0; tile < nTiles; ++tile) {
    const int gBase = start0 + tile * TILE;
    const int base  = gBase + wid * 256;       // this wave's 256-sample block
    const int i0    = base + lane * 8;

    if (tile + 1 < nTiles) __builtin_prefetch(xrow + i0 + TILE, 0, 1);

    float4 xa = *(const float4*)(xrow + i0);
    float4 xb = *(const float4*)(xrow + i0 + 4);

    float s[8];
    s[0] = waveshape((xa.x + dc) * g, t); s[1] = waveshape((xa.y + dc) * g, t);
    s[2] = waveshape((xa.z + dc) * g, t); s[3] = waveshape((xa.w + dc) * g, t);
    s[4] = waveshape((xb.x + dc) * g, t); s[5] = waveshape((xb.y + dc) * g, t);
    s[6] = waveshape((xb.z + dc) * g, t); s[7] = waveshape((xb.w + dc) * g, t);

    // left-neighbor shaped sample (s[-1] == 0 at row start, per reference init state)
    float sL = __shfl(s[7], (lane > 0) ? (lane - 1) : 0, 32);
    if (lane == 0) sL = (base == 0) ? 0.0f : waveshape((xrow[base - 1] + dc) * g, t);

    float d0[8];
    d0[0] = s[0] - sL;
#pragma unroll
    for (int k = 1; k < 8; ++k) d0[k] = s[k] - s[k - 1];

    float4* dp = (float4*)&dbuf[wid][lane * 8];
    dp[0] = make_float4(d0[0], d0[1], d0[2], d0[3]);
    dp[1] = make_float4(d0[4], d0[5], d0[6], d0[7]);
    __syncthreads();

    // Y_local = T x D  via 4 chained fp32 WMMAs (K=4 each).
    // B layout mirror of A: lanes 0-15 -> (K=0,N=lane),(K=1); lanes 16-31 -> (K=2),(K=3).
    v8f acc = {};
#pragma unroll
    for (int kk = 0; kk < 4; ++kk) {
      int p = 16 * j + 4 * kk + (half ? 2 : 0);
      v2f b;
      b.x = dbuf[wid][p];
      b.y = dbuf[wid][p + 1];
      acc = __builtin_amdgcn_wmma_f32_16x16x4_f32(false, afrag[kk], false, b,
                                                  (short)0, acc, false, false);
    }

    // Affine carry scan across the 16 columns: c_j = L_j + alpha^16 * c_{j-1}.
    float Lj = __shfl(acc[7], 16 + j, 32);     // y_local[15][j] lives in VGPR7, lanes 16-31
    float Ai = m16, Bi = Lj;
#pragma unroll
    for (int sft = 1; sft < 16; sft <<= 1) {
      float pA = __shfl(Ai, (lane >= sft) ? (lane - sft) : lane, 32);
      float pB = __shfl(Bi, (lane >= sft) ? (lane - sft) : lane, 32);
      if (j >= sft) { Bi = Bi + Ai * pB; Ai = Ai * pA; }
    }
    float eA = __shfl(Ai, (lane > 0) ? (lane - 1) : 0, 32);
    float eB = __shfl(Bi, (lane > 0) ? (lane - 1) : 0, 32);
    if (j == 0) { eA = 1.0f; eB = 0.0f; }
    float Bw = __shfl(Bi, 15, 32);             // wave-block aggregate (zero carry-in)

    if (lane == 0) waveB[wid] = Bw;
    __syncthreads();

    // Serial affine composition across the 8 waves (redundant per thread, 8 FMAs).
    float cAcc = cMacro, cw = cMacro;
#pragma unroll
    for (int u = 0; u < WAVES; ++u) {
      if (u == wid) cw = cAcc;                 // carry entering this wave's block
      cAcc = waveB[u] + a256 * cAcc;
    }
    cMacro = cAcc;                             // carry into next macro-tile

    float colC = eB + eA * cw;                 // carry entering column j
#pragma unroll
    for (int v = 0; v < 8; ++v) acc[v] += pw[v] * colC;

    // Transpose (M,j) fragment layout -> linear sample order through LDS.
#pragma unroll
    for (int v = 0; v < 8; ++v) ybuf[wid][16 * j + v + 8 * half] = acc[v];
    __syncthreads();

    if (gBase >= outStart) {                   // skip warm-up region
      const float* yp = &ybuf[wid][lane * 8];
      float4 ya = *(const float4*)yp;
      float4 yb = *(const float4*)(yp + 4);
      float4 oa, ob;
      oa.x = om * xa.x + mv * ya.x; oa.y = om * xa.y + mv * ya.y;
      oa.z = om * xa.z + mv * ya.z; oa.w = om * xa.w + mv * ya.w;
      ob.x = om * xb.x + mv * yb.x; ob.y = om * xb.y + mv * yb.y;
      ob.z = om * xb.z + mv * yb.z; ob.w = om * xb.w + mv * yb.w;
      *(float4*)(orow + i0)     = oa;
      *(float4*)(orow + i0 + 4) = ob;
    }
  }
}

extern "C" void kernel_launch(void* const* d_in, const int* in_sizes, int n_in,
                              void* d_out, int out_size, void* d_ws, size_t ws_size,
                              hipStream_t stream) {
  const float* x     = (const float*)d_in[0];
  const float* drive = (const float*)d_in[1];
  const float* bias  = (const float*)d_in[2];
  const float* mix   = (const float*)d_in[3];
  const int*   typ   = (const int*)d_in[4];
  float*       out   = (float*)d_out;

  int rows = in_sizes[1];                      // drive has shape (B,1) -> B elements
  int N    = in_sizes[0] / rows;               // 262144

  int segs = 4;                                // parallel segments per row
  while (segs > 1 && (N % (segs * TILE)) != 0) segs >>= 1;
  int segLen = N / segs;
  int warm = 8192;                             // alpha^8192 ~ 7e-11: below fp32 noise
  if (warm > segLen) warm = segLen;

  dim3 grid(segs, rows);
  dim3 block(256);
  hipLaunchKernelGGL(KHDistortionDSP_hpf_wmma, grid, block, 0, stream,
                     x, drive, bias, mix, typ, out, N, segLen, warm);
}